// GIN_72009421684761
// MI455X (gfx1250) — compile-verified
//
#include <hip/hip_runtime.h>

typedef float v2f __attribute__((ext_vector_type(2)));
typedef float v8f __attribute__((ext_vector_type(8)));

#define HDIM 128  // hidden width (N-dim of all WMMA GEMMs)

// ---------------------------------------------------------------------------
// Fused WMMA f32 GEMM: out[M,128] = f(A)[M,K] @ W[K,128] + bias, with optional
// fused epilogues.  One 256-thread block = 8 waves; W staged in LDS once per
// block; each wave computes a 16x128 strip using V_WMMA_F32_16X16X4_F32.
//   A_OP == 0 : A as-is
//   A_OP == 1 : A + A2                    (h + aggr, GIN eps=0)
//   A_OP == 2 : relu((A - mu)*rsig*g + b) (BatchNorm + ReLU fused into A load)
//   STATS     : atomically accumulate per-column sum / sum^2 of the output
//   OUT_RELU  : relu on output
// ---------------------------------------------------------------------------
template <int A_OP, bool STATS, bool OUT_RELU>
__global__ __launch_bounds__(256) void gemm_wmma_kernel(
    const float* __restrict__ A, const float* __restrict__ A2,
    const float* __restrict__ W, const float* __restrict__ bias,
    const float* __restrict__ mu, const float* __restrict__ rsig,
    const float* __restrict__ gam, const float* __restrict__ bet,
    float* __restrict__ out, float* __restrict__ colsum,
    float* __restrict__ colsumsq, int M, int K)
{
    extern __shared__ float sW[];  // K * 128 floats (<= 64 KB)

    const int tid = threadIdx.x;

    // Cooperative stage of W into LDS (float4 vectorized).
    const int elems4 = (K * HDIM) >> 2;
    const float4* Wv = (const float4*)W;
    float4* sWv = (float4*)sW;
    for (int i = tid; i < elems4; i += 256) sWv[i] = Wv[i];
    __syncthreads();

    const int wave = tid >> 5;
    const int lane = tid & 31;
    const int l    = lane & 15;
    const int hi   = lane >> 4;  // 0: K=k0,k0+1 / rows 0..7 ; 1: K=k0+2,k0+3 / rows 8..15

    const int Mtiles = (M + 15) >> 4;
    const int mtile  = blockIdx.x * 8 + wave;
    if (mtile >= Mtiles) return;  // uniform per-wave exit (EXEC stays all-ones)
    const int m0 = mtile << 4;

    int rowA = m0 + l;
    if (rowA > M - 1) rowA = M - 1;  // clamp for (unused here) tail tiles
    const size_t abase = (size_t)rowA * (size_t)K;

    v8f acc[8];
#pragma unroll
    for (int t = 0; t < 8; ++t)
#pragma unroll
        for (int v = 0; v < 8; ++v) acc[t][v] = 0.0f;

    for (int k = 0; k < K; k += 4) {
        const int ka = k + 2 * hi;

        // ---- A fragment (16x4), one float2 per lane, with fused transform
        v2f a;
        {
            const float2 av = *(const float2*)(A + abase + ka);
            float ax = av.x, ay = av.y;
            if (A_OP == 1) {
                const float2 gv = *(const float2*)(A2 + abase + ka);
                ax += gv.x;
                ay += gv.y;
            } else if (A_OP == 2) {
                const float2 muv = *(const float2*)(mu + ka);
                const float2 rsv = *(const float2*)(rsig + ka);
                const float2 gv  = *(const float2*)(gam + ka);
                const float2 bv  = *(const float2*)(bet + ka);
                ax = fmaxf((ax - muv.x) * rsv.x * gv.x + bv.x, 0.0f);
                ay = fmaxf((ay - muv.y) * rsv.y * gv.y + bv.y, 0.0f);
            }
            a.x = ax;
            a.y = ay;
        }

        // ---- 8 N-tiles reuse the A fragment; B fragments from LDS
#pragma unroll
        for (int t = 0; t < 8; ++t) {
            const int n0 = t << 4;
            v2f b;
            b.x = sW[ka * HDIM + n0 + l];
            b.y = sW[(ka + 1) * HDIM + n0 + l];
            acc[t] = __builtin_amdgcn_wmma_f32_16x16x4_f32(
                false, a, false, b, (short)0, acc[t], false, false);
        }
    }

    // ---- Epilogue: bias (+relu) (+column stats) + store
#pragma unroll
    for (int t = 0; t < 8; ++t) {
        const int n0  = t << 4;
        const int col = n0 + l;
        const float bcol = bias[col];
        float s = 0.0f, ss = 0.0f;
#pragma unroll
        for (int v = 0; v < 8; ++v) {
            const int row = m0 + v + 8 * hi;
            float val = acc[t][v] + bcol;
            if (OUT_RELU) val = fmaxf(val, 0.0f);
            if (row < M) {
                out[(size_t)row * HDIM + col] = val;
                if (STATS) {
                    s += val;
                    ss += val * val;
                }
            }
        }
        if (STATS) {
            atomicAdd(&colsum[col], s);
            atomicAdd(&colsumsq[col], ss);
        }
    }
}

// ---------------------------------------------------------------------------
// Edge message + scatter: one wave per edge, lane handles 4 features.
//   aggr[dst] += relu(h[src] + e[edge])
// ---------------------------------------------------------------------------
__global__ __launch_bounds__(256) void edge_msg_kernel(
    const float* __restrict__ h, const float* __restrict__ e,
    const int* __restrict__ src, const int* __restrict__ dst,
    float* __restrict__ aggr, int E)
{
    const int edge = blockIdx.x * 8 + (threadIdx.x >> 5);
    if (edge >= E) return;
    const int lane = threadIdx.x & 31;
    const int s = src[edge];
    const int d = dst[edge];
    const float4 hv = *(const float4*)(h + (size_t)s * HDIM + lane * 4);
    const float4 ev = *(const float4*)(e + (size_t)edge * HDIM + lane * 4);
    float4 m;
    m.x = fmaxf(hv.x + ev.x, 0.0f);
    m.y = fmaxf(hv.y + ev.y, 0.0f);
    m.z = fmaxf(hv.z + ev.z, 0.0f);
    m.w = fmaxf(hv.w + ev.w, 0.0f);
    float* ap = aggr + (size_t)d * HDIM + lane * 4;
    atomicAdd(ap + 0, m.x);
    atomicAdd(ap + 1, m.y);
    atomicAdd(ap + 2, m.z);
    atomicAdd(ap + 3, m.w);
}

// ---------------------------------------------------------------------------
// BatchNorm stat finalize: mu = sum/N ; rsig = rsqrt(sumsq/N - mu^2 + eps)
// ---------------------------------------------------------------------------
__global__ void bn_finalize_kernel(const float* __restrict__ stats,
                                   float* __restrict__ mu,
                                   float* __restrict__ rsig, float invN)
{
    const int c  = threadIdx.x;
    const float m = stats[c] * invN;
    const float v = stats[HDIM + c] * invN - m * m;
    mu[c]   = m;
    rsig[c] = rsqrtf(v + 1e-5f);
}

// ---------------------------------------------------------------------------
// Global add pool: one wave per node, lane handles 4 features.
// ---------------------------------------------------------------------------
__global__ __launch_bounds__(256) void pool_kernel(
    const float* __restrict__ h, const int* __restrict__ batch,
    float* __restrict__ pooled, int N)
{
    const int node = blockIdx.x * 8 + (threadIdx.x >> 5);
    if (node >= N) return;
    const int lane = threadIdx.x & 31;
    const int g = batch[node];
    const float4 hv = *(const float4*)(h + (size_t)node * HDIM + lane * 4);
    float* pp = pooled + (size_t)g * HDIM + lane * 4;
    atomicAdd(pp + 0, hv.x);
    atomicAdd(pp + 1, hv.y);
    atomicAdd(pp + 2, hv.z);
    atomicAdd(pp + 3, hv.w);
}

// ---------------------------------------------------------------------------
// Final classifier: [G,128] @ [128,C] + b  (tiny: 512x10 threads, scalar)
// ---------------------------------------------------------------------------
__global__ void fc_kernel(const float* __restrict__ pooled,
                          const float* __restrict__ w,
                          const float* __restrict__ b,
                          float* __restrict__ out, int G, int C)
{
    const int idx = blockIdx.x * blockDim.x + threadIdx.x;
    if (idx >= G * C) return;
    const int g = idx / C;
    const int c = idx % C;
    float acc = b[c];
    const float* pg = pooled + (size_t)g * HDIM;
#pragma unroll 8
    for (int k = 0; k < HDIM; ++k) acc += pg[k] * w[k * C + c];
    out[idx] = acc;
}

// ---------------------------------------------------------------------------
extern "C" void kernel_launch(void* const* d_in, const int* in_sizes, int n_in,
                              void* d_out, int out_size, void* d_ws,
                              size_t ws_size, hipStream_t stream)
{
    const float* x         = (const float*)d_in[0];
    const float* edge_attr = (const float*)d_in[1];
    const float* node_w    = (const float*)d_in[2];
    const float* node_b    = (const float*)d_in[3];
    const float* edge_w    = (const float*)d_in[4];
    const float* edge_b    = (const float*)d_in[5];
    const float* lin1_w    = (const float*)d_in[6];
    const float* lin1_b    = (const float*)d_in[7];
    const float* bn_g      = (const float*)d_in[8];
    const float* bn_b      = (const float*)d_in[9];
    const float* lin2_w    = (const float*)d_in[10];
    const float* lin2_b    = (const float*)d_in[11];
    const float* fc_w      = (const float*)d_in[12];
    const float* fc_b      = (const float*)d_in[13];
    const int*   edge_idx  = (const int*)d_in[14];
    const int*   batch     = (const int*)d_in[15];
    (void)n_in; (void)ws_size;

    const int H  = in_sizes[3];           // 128
    const int XD = in_sizes[2] / H;       // 64
    const int N  = in_sizes[0] / XD;      // 50000
    const int ED = in_sizes[4] / H;       // 16
    const int E  = in_sizes[1] / ED;      // 640000
    const int C  = in_sizes[13];          // 10
    const int G  = out_size / C;          // 512
    const int L  = in_sizes[7] / H;       // 3

    // Workspace carve-out (256B aligned slices)
    char* p = (char*)d_ws;
    auto take = [&](size_t bytes) -> char* {
        char* r = p;
        p += (bytes + 255) & ~(size_t)255;
        return r;
    };
    float* h      = (float*)take((size_t)N * H * 4);
    float* z      = (float*)take((size_t)N * H * 4);
    float* aggr   = (float*)take((size_t)N * H * 4);
    float* eenc   = (float*)take((size_t)E * H * 4);
    float* pooled = (float*)take((size_t)G * H * 4);
    float* stats  = (float*)take((size_t)4 * H * 4);  // colsum | colsumsq | mu | rsig
    float* mu     = stats + 2 * H;
    float* rsig   = stats + 3 * H;

    const int nstrips = ((N + 15) / 16 + 7) / 8;
    const int estrips = ((E + 15) / 16 + 7) / 8;

    // Node encoder: h = x @ node_w + node_b
    gemm_wmma_kernel<0, false, false><<<nstrips, 256, (size_t)XD * H * 4, stream>>>(
        x, nullptr, node_w, node_b, nullptr, nullptr, nullptr, nullptr,
        h, nullptr, nullptr, N, XD);

    // Edge encoder: e = edge_attr @ edge_w + edge_b
    gemm_wmma_kernel<0, false, false><<<estrips, 256, (size_t)ED * H * 4, stream>>>(
        edge_attr, nullptr, edge_w, edge_b, nullptr, nullptr, nullptr, nullptr,
        eenc, nullptr, nullptr, E, ED);

    for (int i = 0; i < L; ++i) {
        // aggr = segment_sum(relu(h[src] + e), dst)
        hipMemsetAsync(aggr, 0, (size_t)N * H * 4, stream);
        edge_msg_kernel<<<(E + 7) / 8, 256, 0, stream>>>(
            h, eenc, edge_idx, edge_idx + E, aggr, E);

        // z = (h + aggr) @ lin1 + b1, accumulating per-column sum/sum^2
        hipMemsetAsync(stats, 0, (size_t)2 * H * 4, stream);
        gemm_wmma_kernel<1, true, false><<<nstrips, 256, (size_t)H * H * 4, stream>>>(
            h, aggr, lin1_w + (size_t)i * H * H, lin1_b + (size_t)i * H,
            nullptr, nullptr, nullptr, nullptr, z, stats, stats + H, N, H);

        bn_finalize_kernel<<<1, H, 0, stream>>>(stats, mu, rsig, 1.0f / (float)N);

        // h = relu( relu(BN(z)) @ lin2 + b2 )
        gemm_wmma_kernel<2, false, true><<<nstrips, 256, (size_t)H * H * 4, stream>>>(
            z, nullptr, lin2_w + (size_t)i * H * H, lin2_b + (size_t)i * H,
            mu, rsig, bn_g + (size_t)i * H, bn_b + (size_t)i * H,
            h, nullptr, nullptr, N, H);
    }

    // Global add pool + classifier
    hipMemsetAsync(pooled, 0, (size_t)G * H * 4, stream);
    pool_kernel<<<(N + 7) / 8, 256, 0, stream>>>(h, batch, pooled, N);
    fc_kernel<<<(G * C + 255) / 256, 256, 0, stream>>>(
        pooled, fc_w, fc_b, (float*)d_out, G, C);
}